// Low_Rank_HORNN_76020921140341
// MI455X (gfx1250) — compile-verified
//
#include <hip/hip_runtime.h>
#include <hip/hip_bf16.h>
#include <math.h>

// Problem constants (from reference)
#define BB   64
#define TT   1024
#define INN  64
#define OUTN 32
#define HH   512

typedef float v2f __attribute__((ext_vector_type(2)));
typedef float v8f __attribute__((ext_vector_type(8)));

// ---------------------------------------------------------------------------
// Kernel 1: input projection  I[b,t,h] = sum_i u[b,t,i]*W_in[h,i] + b_in[h]
// f32 WMMA 16x16x4. Rows = b*T+t (65536), cols = h (512), K = 64.
// Each wave: one 16-row tile x FOUR 16-col tiles (4 accumulators), so one A
// load feeds 4 WMMAs; the 4 B tiles are immediate offsets off one pointer.
// Output written into the traj region of d_out in [B,T,H] layout.
// ---------------------------------------------------------------------------
__global__ __launch_bounds__(256) void hornn_input_proj(
    const float* __restrict__ u, const float* __restrict__ W_in,
    const float* __restrict__ b_in, float* __restrict__ I)
{
    const int CG = 4;                              // col tiles per wave
    const int COL_GROUPS = HH / (16 * CG);         // 8
    int wave = threadIdx.x >> 5;
    int lane = threadIdx.x & 31;
    int tile = blockIdx.x * 8 + wave;              // global wave-tile id
    int rowTile  = tile / COL_GROUPS;
    int colGroup = tile % COL_GROUPS;
    int row0 = rowTile * 16;
    int h0   = colGroup * 16 * CG;

    int lhalf = lane & 15;
    int hi    = (lane >> 4) & 1;                   // 0: lanes 0-15, 1: lanes 16-31
    int arow  = row0 + lhalf;                      // A: lane indexes M

    v8f acc[CG] = {};
    const float* __restrict__ urow = u    + (long)arow * INN;
    const float* __restrict__ wrow = W_in + (long)(h0 + lhalf) * INN;  // col tile c at +c*16*64

#pragma unroll 4
    for (int k = 0; k < INN; k += 4) {
        int ka = k + (hi ? 2 : 0);
        v2f a;
        a.x = urow[ka];  a.y = urow[ka + 1];       // A[M, K0/K1] (lo) or K2/K3 (hi)
#pragma unroll
        for (int c = 0; c < CG; ++c) {
            v2f b;
            b.x = wrow[ka + c * 16 * INN];
            b.y = wrow[ka + c * 16 * INN + 1];
            acc[c] = __builtin_amdgcn_wmma_f32_16x16x4_f32(
                false, a, false, b, (short)0, acc[c], false, false);
        }
    }

#pragma unroll
    for (int c = 0; c < CG; ++c) {
        float bias = b_in[h0 + c * 16 + lhalf];
#pragma unroll
        for (int j = 0; j < 8; ++j) {
            int M = j + (hi ? 8 : 0);
            I[(long)(row0 + M) * HH + h0 + c * 16 + lhalf] = acc[c][j] + bias;
        }
    }
}

// ---------------------------------------------------------------------------
// Kernel 2: the sequential scan. One block per batch (64 blocks, 256 threads).
// Each thread owns h = tid and tid+256; state x lives in registers.
// Rank-2 reductions: wave32 shuffle reduce + 1-barrier double-buffered LDS.
// traj region already holds I[b,t,h]; it is read then overwritten with x_new.
// ---------------------------------------------------------------------------
__device__ inline float wave_reduce(float v) {
#pragma unroll
    for (int off = 16; off > 0; off >>= 1)
        v += __shfl_xor(v, off, 32);
    return v;
}

__global__ __launch_bounds__(256) void hornn_scan(
    const float* __restrict__ x0, const float* __restrict__ noise,
    const float* __restrict__ M_rnn, const float* __restrict__ N_rnn,
    const float* __restrict__ L_tb, const float* __restrict__ M_tb,
    const float* __restrict__ N_tb,
    float* __restrict__ traj, float* __restrict__ xlast)
{
    const float TAU = 0.2f, NOISE_STD = 0.05f;
    const float inv_h  = 1.0f / (float)HH;
    const float inv_h2 = inv_h * inv_h;

    int b   = blockIdx.x;
    int tid = threadIdx.x;
    int lane = tid & 31, wv = tid >> 5;
    int h[2] = { tid, tid + 256 };

    // Per-thread weights for the owned h rows ([H,2] layout)
    float wN[2][2], wMt[2][2], wNt[2][2], wMr[2][2], wL[2][2], x[2];
#pragma unroll
    for (int i = 0; i < 2; ++i) {
        int hh = h[i];
        wN[i][0]  = N_rnn[hh*2+0];  wN[i][1]  = N_rnn[hh*2+1];
        wMt[i][0] = M_tb [hh*2+0];  wMt[i][1] = M_tb [hh*2+1];
        wNt[i][0] = N_tb [hh*2+0];  wNt[i][1] = N_tb [hh*2+1];
        wMr[i][0] = M_rnn[hh*2+0];  wMr[i][1] = M_rnn[hh*2+1];
        wL[i][0]  = L_tb [hh*2+0];  wL[i][1]  = L_tb [hh*2+1];
        x[i]      = x0[(long)b * HH + hh];
    }

    float* __restrict__ trajb = traj + (long)b * TT * HH;     // [T,H] slice
    const float* __restrict__ noiseb = noise + (long)b * HH;  // noise[t,b,h]

    __shared__ float part[2][8][6];                           // double-buffered

    for (int t = 0; t < TT; ++t) {
        // prefetch next step's operands into cache (global_prefetch_b8)
        if (t + 1 < TT) {
            __builtin_prefetch(&noiseb[(long)(t + 1) * BB * HH + h[0]], 0, 0);
            __builtin_prefetch(&noiseb[(long)(t + 1) * BB * HH + h[1]], 0, 0);
            __builtin_prefetch(&trajb[(long)(t + 1) * HH + h[0]], 0, 0);
            __builtin_prefetch(&trajb[(long)(t + 1) * HH + h[1]], 0, 0);
        }

        float r0 = tanhf(x[0]), r1 = tanhf(x[1]);

        // 6 rank-space partial sums
        float s0 = r0*wN[0][0]  + r1*wN[1][0];
        float s1 = r0*wN[0][1]  + r1*wN[1][1];
        float s2 = r0*wMt[0][0] + r1*wMt[1][0];
        float s3 = r0*wMt[0][1] + r1*wMt[1][1];
        float s4 = r0*wNt[0][0] + r1*wNt[1][0];
        float s5 = r0*wNt[0][1] + r1*wNt[1][1];

        s0 = wave_reduce(s0); s1 = wave_reduce(s1); s2 = wave_reduce(s2);
        s3 = wave_reduce(s3); s4 = wave_reduce(s4); s5 = wave_reduce(s5);

        int buf = t & 1;
        if (lane == 0) {
            part[buf][wv][0] = s0; part[buf][wv][1] = s1; part[buf][wv][2] = s2;
            part[buf][wv][3] = s3; part[buf][wv][4] = s4; part[buf][wv][5] = s5;
        }
        __syncthreads();   // single barrier per step (double buffering)

        float pN0 = 0.f, pN1 = 0.f, pM0 = 0.f, pM1 = 0.f, pQ0 = 0.f, pQ1 = 0.f;
#pragma unroll
        for (int w = 0; w < 8; ++w) {
            pN0 += part[buf][w][0]; pN1 += part[buf][w][1];
            pM0 += part[buf][w][2]; pM1 += part[buf][w][3];
            pQ0 += part[buf][w][4]; pQ1 += part[buf][w][5];
        }
        float q0 = pM0 * pQ0, q1 = pM1 * pQ1;

#pragma unroll
        for (int i = 0; i < 2; ++i) {
            int hh = h[i];
            float h_rnn = (pN0 * wMr[i][0] + pN1 * wMr[i][1]) * inv_h;
            float h_tb  = (q0  * wL[i][0]  + q1  * wL[i][1])  * inv_h2;
            float I_t   = trajb[(long)t * HH + hh];               // read I
            float n_t   = noiseb[(long)t * BB * HH + hh];
            float rec   = h_rnn + h_tb + I_t;
            float xn    = x[i] + NOISE_STD * n_t + TAU * (rec - x[i]);
            x[i] = xn;
            trajb[(long)t * HH + hh] = xn;                        // overwrite with traj
        }
    }

#pragma unroll
    for (int i = 0; i < 2; ++i)
        xlast[(long)b * HH + h[i]] = x[i];
}

// ---------------------------------------------------------------------------
// Kernel 3: output projection  out[b,t,o] = tanh(traj[b,t,:]) . W_out[o,:] + b_out[o]
// f32 WMMA 16x16x4. Rows = b*T+t (65536), cols = o (32), K = 512.
// Each wave handles BOTH 16-col tiles (2 accumulators): one tanh'd A load
// feeds 2 WMMAs -> traj read once, tanh evaluated once per element.
// ---------------------------------------------------------------------------
__global__ __launch_bounds__(256) void hornn_output_proj(
    const float* __restrict__ traj, const float* __restrict__ W_out,
    const float* __restrict__ b_out, float* __restrict__ out)
{
    const int CG = OUTN / 16;                      // 2 col tiles per wave
    int wave = threadIdx.x >> 5;
    int lane = threadIdx.x & 31;
    int rowTile = blockIdx.x * 8 + wave;           // one wave == one row tile
    int row0 = rowTile * 16;

    int lhalf = lane & 15;
    int hi    = (lane >> 4) & 1;
    int arow  = row0 + lhalf;

    v8f acc[CG] = {};
    const float* __restrict__ trow = traj  + (long)arow * HH;
    const float* __restrict__ wrow = W_out + (long)lhalf * HH;   // col tile c at +c*16*512

    for (int k = 0; k < HH; k += 4) {
        int ka = k + (hi ? 2 : 0);
        v2f a;
        a.x = tanhf(trow[ka]);  a.y = tanhf(trow[ka + 1]);
#pragma unroll
        for (int c = 0; c < CG; ++c) {
            v2f b;
            b.x = wrow[ka + c * 16 * HH];
            b.y = wrow[ka + c * 16 * HH + 1];
            acc[c] = __builtin_amdgcn_wmma_f32_16x16x4_f32(
                false, a, false, b, (short)0, acc[c], false, false);
        }
    }

#pragma unroll
    for (int c = 0; c < CG; ++c) {
        float bias = b_out[c * 16 + lhalf];
#pragma unroll
        for (int j = 0; j < 8; ++j) {
            int M = j + (hi ? 8 : 0);
            out[(long)(row0 + M) * OUTN + c * 16 + lhalf] = acc[c][j] + bias;
        }
    }
}

// ---------------------------------------------------------------------------
extern "C" void kernel_launch(void* const* d_in, const int* in_sizes, int n_in,
                              void* d_out, int out_size, void* d_ws, size_t ws_size,
                              hipStream_t stream) {
    const float* u     = (const float*)d_in[0];
    const float* x0    = (const float*)d_in[1];
    const float* noise = (const float*)d_in[2];
    const float* W_in  = (const float*)d_in[3];
    const float* b_in  = (const float*)d_in[4];
    const float* M_rnn = (const float*)d_in[5];
    const float* N_rnn = (const float*)d_in[6];
    const float* L_tb  = (const float*)d_in[7];
    const float* M_tb  = (const float*)d_in[8];
    const float* N_tb  = (const float*)d_in[9];
    const float* W_out = (const float*)d_in[10];
    const float* b_out = (const float*)d_in[11];

    float* out   = (float*)d_out;                       // [B,T,OUT]
    float* xlast = out + (long)BB * TT * OUTN;          // [B,H]
    float* traj  = xlast + (long)BB * HH;               // [B,T,H] (also holds I)

    // 1) I = u @ W_in^T + b_in  ->  traj region, [B,T,H]
    {
        int waveTiles = (BB * TT / 16) * (HH / 64);     // 4096 * 8 = 32768
        hornn_input_proj<<<waveTiles / 8, 256, 0, stream>>>(u, W_in, b_in, traj);
    }
    // 2) sequential scan, state in registers, one block per batch
    hornn_scan<<<BB, 256, 0, stream>>>(x0, noise, M_rnn, N_rnn, L_tb, M_tb, N_tb,
                                       traj, xlast);
    // 3) out = tanh(traj) @ W_out^T + b_out
    {
        int rowTiles = BB * TT / 16;                    // 4096 waves, 512 blocks
        hornn_output_proj<<<rowTiles / 8, 256, 0, stream>>>(traj, W_out, b_out, out);
    }
}